// SEALGIN_53420803228462
// MI455X (gfx1250) — compile-verified
//
#include <hip/hip_runtime.h>

#define N_NODES  50000
#define N_EDGES  600000
#define HIDDEN   128
#define LAYERS   3
#define N_GRAPHS 512
#define XCAT     (LAYERS * HIDDEN) /* 384 */
#define BN_EPS   1e-5f

typedef float v2f __attribute__((ext_vector_type(2)));
typedef float v8f __attribute__((ext_vector_type(8)));

// ---------------------------------------------------------------- utilities

__global__ void zero_kernel(float* __restrict__ p, int n) {
    int gid = blockIdx.x * blockDim.x + threadIdx.x;
    if (gid < n) p[gid] = 0.0f;
}

// x[n][c] = z_table[z[n]][c], float4 per thread
__global__ void embed_kernel(const int* __restrict__ z,
                             const float* __restrict__ table,
                             float* __restrict__ x) {
    int gid = blockIdx.x * blockDim.x + threadIdx.x;  // N_NODES * 32
    if (gid >= N_NODES * 32) return;
    int n  = gid >> 5;
    int c4 = (gid & 31) * 4;
    float4 v = *(const float4*)(table + (size_t)z[n] * HIDDEN + c4);
    *(float4*)(x + (size_t)n * HIDDEN + c4) = v;
}

// agg[dst[e]][c] += x[src[e]][c]  (f32 atomics resolve in L2; agg fits in L2)
__global__ void scatter_kernel(const int* __restrict__ src,
                               const int* __restrict__ dst,
                               const float* __restrict__ x,
                               float* __restrict__ agg) {
    long gid = (long)blockIdx.x * blockDim.x + threadIdx.x;  // N_EDGES * 32
    if (gid >= (long)N_EDGES * 32) return;
    int e  = (int)(gid >> 5);
    int c4 = (int)(gid & 31) * 4;
    int s = src[e], d = dst[e];
    float4 v = *(const float4*)(x + (size_t)s * HIDDEN + c4);
    float* ap = agg + (size_t)d * HIDDEN + c4;
    atomicAdd(ap + 0, v.x);
    atomicAdd(ap + 1, v.y);
    atomicAdd(ap + 2, v.z);
    atomicAdd(ap + 3, v.w);
}

// ------------------------------------------------- fused GIN layer (WMMA f32)
// Block = 16 nodes x 128 features, 8 waves; wave w -> output cols [16w,16w+16).
// h1 = relu((x+agg) @ W1 + b1); h2 = relu(h1 @ W2 + b2); out = BN(h2)
// Writes x in place (block owns its 16 rows) and the xcat slice.
__global__ __launch_bounds__(256) void gin_layer_kernel(
    float* __restrict__ x, const float* __restrict__ agg,
    const float* __restrict__ W1, const float* __restrict__ b1,
    const float* __restrict__ W2, const float* __restrict__ b2,
    const float* __restrict__ bng, const float* __restrict__ bnb,
    const float* __restrict__ bnm, const float* __restrict__ bnv,
    float* __restrict__ xcat, int layer)
{
    __shared__ float sA[16][HIDDEN + 4];  // +4 pad: A-frag reads hit distinct banks
    __shared__ float sH[16][HIDDEN + 4];

    const int node0 = blockIdx.x * 16;
    const int tid   = threadIdx.x;

    // stage A tile: x + agg
    for (int i = tid; i < 16 * HIDDEN; i += 256) {
        int r = i >> 7, c = i & (HIDDEN - 1);
        size_t off = (size_t)(node0 + r) * HIDDEN + c;
        sA[r][c] = x[off] + agg[off];
    }
    __syncthreads();

    const int wave  = tid >> 5;
    const int lane  = tid & 31;
    const int nbase = wave * 16;
    const int col   = nbase + (lane & 15);   // output column (N)
    const int arow  = lane & 15;             // A row for this lane
    const int khalf = (lane >> 4) * 2;       // lanes 16-31 carry K+2,K+3
    const int mbase = (lane < 16) ? 0 : 8;   // C/D: VGPR r -> M=r or M=r+8

    // ---- GEMM 1: (x+agg) @ W1 + b1, ReLU
    v8f acc;
    {
        float bias = b1[col];
        #pragma unroll
        for (int i = 0; i < 8; i++) acc[i] = bias;
    }
    #pragma unroll 4
    for (int k = 0; k < HIDDEN; k += 4) {
        int kb = k + khalf;
        v2f a; a.x = sA[arow][kb];            a.y = sA[arow][kb + 1];
        v2f b; b.x = W1[kb * HIDDEN + col];   b.y = W1[(kb + 1) * HIDDEN + col];
        acc = __builtin_amdgcn_wmma_f32_16x16x4_f32(false, a, false, b,
                                                    (short)0, acc, false, false);
    }
    #pragma unroll
    for (int i = 0; i < 8; i++) {
        float v = acc[i] > 0.0f ? acc[i] : 0.0f;
        sH[mbase + i][col] = v;
    }
    __syncthreads();

    // ---- GEMM 2: h1 @ W2 + b2, ReLU, BatchNorm (inference)
    v8f acc2;
    {
        float bias = b2[col];
        #pragma unroll
        for (int i = 0; i < 8; i++) acc2[i] = bias;
    }
    #pragma unroll 4
    for (int k = 0; k < HIDDEN; k += 4) {
        int kb = k + khalf;
        v2f a; a.x = sH[arow][kb];            a.y = sH[arow][kb + 1];
        v2f b; b.x = W2[kb * HIDDEN + col];   b.y = W2[(kb + 1) * HIDDEN + col];
        acc2 = __builtin_amdgcn_wmma_f32_16x16x4_f32(false, a, false, b,
                                                     (short)0, acc2, false, false);
    }
    {
        float scale = bng[col] * rsqrtf(bnv[col] + BN_EPS);
        float shift = bnb[col] - bnm[col] * scale;
        #pragma unroll
        for (int i = 0; i < 8; i++) {
            float v  = acc2[i] > 0.0f ? acc2[i] : 0.0f;
            float bn = v * scale + shift;
            int node = node0 + mbase + i;
            x[(size_t)node * HIDDEN + col] = bn;                       // next-layer input
            xcat[(size_t)node * XCAT + layer * HIDDEN + col] = bn;     // concat slice
        }
    }
}

// ---------------------------------------------------------------- pooling

__global__ void pool_kernel(const float* __restrict__ xcat,
                            const int* __restrict__ batch,
                            float* __restrict__ sums) {
    int gid = blockIdx.x * blockDim.x + threadIdx.x;  // N_NODES * 96
    if (gid >= N_NODES * 96) return;
    int n  = gid / 96;
    int c4 = (gid % 96) * 4;
    int g  = batch[n];
    float4 v = *(const float4*)(xcat + (size_t)n * XCAT + c4);
    float* sp = sums + (size_t)g * XCAT + c4;
    atomicAdd(sp + 0, v.x);
    atomicAdd(sp + 1, v.y);
    atomicAdd(sp + 2, v.z);
    atomicAdd(sp + 3, v.w);
}

__global__ void count_kernel(const int* __restrict__ batch,
                             float* __restrict__ cnts) {
    int n = blockIdx.x * blockDim.x + threadIdx.x;
    if (n < N_NODES) atomicAdd(&cnts[batch[n]], 1.0f);
}

// ------------------------------------- MLP1: relu(pooled @ lin1_w + b), WMMA
__global__ __launch_bounds__(256) void mlp1_kernel(
    const float* __restrict__ sums, const float* __restrict__ cnts,
    const float* __restrict__ W, const float* __restrict__ b,
    float* __restrict__ hout)
{
    __shared__ float sA[16][XCAT + 4];
    const int g0  = blockIdx.x * 16;   // 32 tiles of 16 graphs
    const int tid = threadIdx.x;

    for (int i = tid; i < 16 * XCAT; i += 256) {
        int r = i / XCAT, c = i % XCAT;
        float cnt = cnts[g0 + r];
        cnt = cnt > 1.0f ? cnt : 1.0f;
        sA[r][c] = sums[(size_t)(g0 + r) * XCAT + c] / cnt;  // mean fold-in
    }
    __syncthreads();

    const int wave  = tid >> 5;
    const int lane  = tid & 31;
    const int col   = wave * 16 + (lane & 15);
    const int arow  = lane & 15;
    const int khalf = (lane >> 4) * 2;
    const int mbase = (lane < 16) ? 0 : 8;

    v8f acc;
    {
        float bias = b[col];
        #pragma unroll
        for (int i = 0; i < 8; i++) acc[i] = bias;
    }
    #pragma unroll 4
    for (int k = 0; k < XCAT; k += 4) {
        int kb = k + khalf;
        v2f a;  a.x  = sA[arow][kb];           a.y  = sA[arow][kb + 1];
        v2f bb; bb.x = W[kb * HIDDEN + col];   bb.y = W[(kb + 1) * HIDDEN + col];
        acc = __builtin_amdgcn_wmma_f32_16x16x4_f32(false, a, false, bb,
                                                    (short)0, acc, false, false);
    }
    #pragma unroll
    for (int i = 0; i < 8; i++) {
        float v = acc[i] > 0.0f ? acc[i] : 0.0f;
        hout[(size_t)(g0 + mbase + i) * HIDDEN + col] = v;
    }
}

// ---------------------------------------------------------------- final dot
__global__ void final_kernel(const float* __restrict__ h,
                             const float* __restrict__ w2,
                             const float* __restrict__ b2,
                             float* __restrict__ out) {
    int g = blockIdx.x * blockDim.x + threadIdx.x;
    if (g >= N_GRAPHS) return;
    float acc = b2[0];
    #pragma unroll 4
    for (int c = 0; c < HIDDEN; c++) acc += h[(size_t)g * HIDDEN + c] * w2[c];
    out[g] = acc;
}

// ---------------------------------------------------------------- launcher

extern "C" void kernel_launch(void* const* d_in, const int* in_sizes, int n_in,
                              void* d_out, int out_size, void* d_ws, size_t ws_size,
                              hipStream_t stream) {
    const int*   z     = (const int*)d_in[0];
    const int*   eidx  = (const int*)d_in[1];
    const int*   batch = (const int*)d_in[2];
    const float* table = (const float*)d_in[3];
    const float* W1s   = (const float*)d_in[4];
    const float* b1s   = (const float*)d_in[5];
    const float* W2s   = (const float*)d_in[6];
    const float* b2s   = (const float*)d_in[7];
    const float* bng   = (const float*)d_in[8];
    const float* bnb   = (const float*)d_in[9];
    const float* bnm   = (const float*)d_in[10];
    const float* bnv   = (const float*)d_in[11];
    const float* l1w   = (const float*)d_in[12];
    const float* l1b   = (const float*)d_in[13];
    const float* l2w   = (const float*)d_in[14];
    const float* l2b   = (const float*)d_in[15];
    float* out = (float*)d_out;

    // workspace partition (floats): ~129 MB total
    float* ws   = (float*)d_ws;
    float* x    = ws;  ws += (size_t)N_NODES * HIDDEN;
    float* agg  = ws;  ws += (size_t)N_NODES * HIDDEN;
    float* xcat = ws;  ws += (size_t)N_NODES * XCAT;
    float* sums = ws;  ws += (size_t)N_GRAPHS * XCAT;   // cnts directly follows
    float* cnts = ws;  ws += N_GRAPHS;
    float* hbuf = ws;  ws += (size_t)N_GRAPHS * HIDDEN;

    const int T = 256;
    const int* src = eidx;
    const int* dst = eidx + N_EDGES;

    embed_kernel<<<(N_NODES * 32 + T - 1) / T, T, 0, stream>>>(z, table, x);
    // zero pooled sums + counts (contiguous) every call (atomics accumulate)
    zero_kernel<<<(N_GRAPHS * XCAT + N_GRAPHS + T - 1) / T, T, 0, stream>>>(
        sums, N_GRAPHS * XCAT + N_GRAPHS);

    for (int l = 0; l < LAYERS; l++) {
        zero_kernel<<<(N_NODES * HIDDEN + T - 1) / T, T, 0, stream>>>(
            agg, N_NODES * HIDDEN);
        scatter_kernel<<<(N_EDGES * 32 + T - 1) / T, T, 0, stream>>>(
            src, dst, x, agg);
        gin_layer_kernel<<<N_NODES / 16, T, 0, stream>>>(
            x, agg,
            W1s + (size_t)l * HIDDEN * HIDDEN, b1s + l * HIDDEN,
            W2s + (size_t)l * HIDDEN * HIDDEN, b2s + l * HIDDEN,
            bng + l * HIDDEN, bnb + l * HIDDEN,
            bnm + l * HIDDEN, bnv + l * HIDDEN,
            xcat, l);
    }

    pool_kernel<<<(N_NODES * 96 + T - 1) / T, T, 0, stream>>>(xcat, batch, sums);
    count_kernel<<<(N_NODES + T - 1) / T, T, 0, stream>>>(batch, cnts);
    mlp1_kernel<<<N_GRAPHS / 16, T, 0, stream>>>(sums, cnts, l1w, l1b, hbuf);
    final_kernel<<<(N_GRAPHS + T - 1) / T, T, 0, stream>>>(hbuf, l2w, l2b, out);
}